// MeanAggregator_17918603558960
// MI455X (gfx1250) — compile-verified
//
#include <hip/hip_runtime.h>
#include <math.h>

// Problem constants (from reference)
#define N_NODES 100000
#define DIM     256          // DIM_IN == DIM_OUT
#define OUTW    512          // 2*DIM_OUT
#define EPSF    1e-9f

typedef __attribute__((ext_vector_type(2))) float v2f;
typedef __attribute__((ext_vector_type(8))) float v8f;

// ---------------------------------------------------------------------------
// Kernel 1: zero the output buffer (its upper half is the SpMM accumulator).
// Exact-size launch: N_NODES*OUTW/4 float4 elements = 12,800,000.
// ---------------------------------------------------------------------------
__global__ __launch_bounds__(256) void zero_out_kernel(float4* __restrict__ out) {
    size_t i = (size_t)blockIdx.x * blockDim.x + threadIdx.x;
    out[i] = make_float4(0.f, 0.f, 0.f, 0.f);
}

// ---------------------------------------------------------------------------
// Kernel 2: edge scatter (SpMM). One wave32 per edge; lane l handles the 8
// contiguous dims [8l, 8l+8). Accumulates into out[row*512 + 256 + d] using
// hardware global_atomic_add_f32. vecs (102 MB) is L2-resident on MI455X.
// ---------------------------------------------------------------------------
__global__ __launch_bounds__(256) void edge_scatter_kernel(
        const float* __restrict__ vecs,
        const int*   __restrict__ rows,
        const int*   __restrict__ cols,
        const float* __restrict__ vals,
        float*       __restrict__ out,
        int E) {
    int lane = threadIdx.x & 31;
    int e = blockIdx.x * (blockDim.x >> 5) + (threadIdx.x >> 5);
    if (e >= E) return;
    int r = rows[e];
    int c = cols[e];
    float v = vals[e];

    const float4* src = (const float4*)(vecs + (size_t)c * DIM);
    float4 a = src[lane * 2 + 0];
    float4 b = src[lane * 2 + 1];

    float* dst = out + (size_t)r * OUTW + DIM + lane * 8;
    unsafeAtomicAdd(dst + 0, v * a.x);
    unsafeAtomicAdd(dst + 1, v * a.y);
    unsafeAtomicAdd(dst + 2, v * a.z);
    unsafeAtomicAdd(dst + 3, v * a.w);
    unsafeAtomicAdd(dst + 4, v * b.x);
    unsafeAtomicAdd(dst + 5, v * b.y);
    unsafeAtomicAdd(dst + 6, v * b.z);
    unsafeAtomicAdd(dst + 7, v * b.w);
}

// ---------------------------------------------------------------------------
// Kernel 3: fused dual-GEMM + layer-norm + ReLU.
// Block = 256 threads (8 waves) handles 16 node rows.
//   - A tiles (self = vecs, neigh = accumulated half of out) staged in LDS,
//     row stride padded to 260 floats for conflict-free ds_load_b64.
//   - Each wave computes 4 of the 32 16x16 output col-tiles with
//     V_WMMA_F32_16X16X4_F32 (K-loop of 64).
//   - Results staged in LDS; per-row mean/var via __shfl_xor butterfly;
//     scale/offset/ReLU applied on the way out.
// ISA f32 16x16x4 layouts (CDNA5 ISA):
//   A: lane L holds M=L%16, K = (L/16)*2 + v          (2 VGPRs)
//   B: lane L holds N=L%16, K = (L/16)*2 + v          (2 VGPRs)
//   C/D: lane L, VGPR v holds M = v + 8*(L/16), N=L%16 (8 VGPRs)
// ---------------------------------------------------------------------------
#define A_STRIDE 260   // padded row stride (floats): 260*4B rows, banks 4 apart

__global__ __launch_bounds__(256) void gemm_norm_kernel(
        const float* __restrict__ vecs,
        const float* __restrict__ W_neigh,
        const float* __restrict__ W_self,
        const float* __restrict__ offset,
        const float* __restrict__ scale,
        float*       __restrict__ out) {
    __shared__ float sSelf [16 * A_STRIDE];
    __shared__ float sNeigh[16 * A_STRIDE];
    __shared__ float sOut  [16 * OUTW];

    const int tid  = threadIdx.x;
    const int row0 = blockIdx.x * 16;

    // ---- Stage both 16x256 A tiles into LDS (float4, 16B-aligned rows) ----
    #pragma unroll
    for (int i = 0; i < 4; ++i) {
        int v4 = tid + i * 256;          // float4 index 0..1023
        int r  = v4 >> 6;                // row 0..15
        int c4 = v4 & 63;                // float4 col 0..63
        float4 s = ((const float4*)(vecs + (size_t)(row0 + r) * DIM))[c4];
        float4 n = ((const float4*)(out  + (size_t)(row0 + r) * OUTW + DIM))[c4];
        *(float4*)&sSelf [r * A_STRIDE + c4 * 4] = s;
        *(float4*)&sNeigh[r * A_STRIDE + c4 * 4] = n;
    }
    __syncthreads();

    const int lane = tid & 31;
    const int wave = tid >> 5;
    const int m  = lane & 15;          // M (A) / N (B) index
    const int kh = (lane >> 4) * 2;    // K sub-offset for this half-wave

    // ---- 8 waves x 4 tiles = 32 output col-tiles (0..15 self, 16..31 neigh)
    for (int tt = 0; tt < 4; ++tt) {
        const int t = wave + tt * 8;
        const float* __restrict__ A = (t < 16) ? sSelf : sNeigh;
        const float* __restrict__ W = (t < 16) ? W_self : W_neigh;
        const int n0 = (t * 16) & (DIM - 1);       // W column base

        const float* Ap = A + m * A_STRIDE + kh;   // LDS: A[m][kh + k0 + v]
        const float* Wp = W + n0 + m;              // global: W[k][n0+m]

        v8f acc = {};
        #pragma unroll 4
        for (int k0 = 0; k0 < DIM; k0 += 4) {
            v2f a; a[0] = Ap[k0];     a[1] = Ap[k0 + 1];          // ds_load_b64
            v2f b; b[0] = Wp[(size_t)(k0 + kh) * DIM];
                   b[1] = Wp[(size_t)(k0 + kh + 1) * DIM];
            acc = __builtin_amdgcn_wmma_f32_16x16x4_f32(
                    /*neg_a=*/false, a, /*neg_b=*/false, b,
                    /*c_mod=*/(short)0, acc,
                    /*reuse_a=*/false, /*reuse_b=*/false);
        }

        // D layout -> LDS out tile (output col base is t*16 for both halves)
        const int c0 = t * 16;
        const int mh = (lane >> 4) * 8;
        #pragma unroll
        for (int v = 0; v < 8; ++v)
            sOut[(mh + v) * OUTW + c0 + m] = acc[v];
    }
    __syncthreads();

    // ---- Per-row layer-norm + scale/offset + ReLU -------------------------
    const int r = tid >> 4;            // row within tile, 0..15
    const int j = tid & 15;            // 16 threads cooperate per row
    float sum = 0.f, sumsq = 0.f;
    #pragma unroll
    for (int k = 0; k < 32; ++k) {
        float x = sOut[r * OUTW + j + k * 16];
        sum   += x;
        sumsq += x * x;
    }
    // butterfly reduction over the 16 threads of this row (within half-wave)
    #pragma unroll
    for (int msk = 8; msk >= 1; msk >>= 1) {
        sum   += __shfl_xor(sum,   msk, 16);
        sumsq += __shfl_xor(sumsq, msk, 16);
    }
    const float mean = sum * (1.f / OUTW);
    const float var  = sumsq * (1.f / OUTW) - mean * mean;
    const float inv  = rsqrtf(var + EPSF);

    float* orow = out + (size_t)(row0 + r) * OUTW;
    #pragma unroll
    for (int k = 0; k < 32; ++k) {
        int c = j + k * 16;
        float x = (sOut[r * OUTW + c] - mean) * inv * scale[c] + offset[c];
        orow[c] = fmaxf(x, 0.f);
    }
}

// ---------------------------------------------------------------------------
extern "C" void kernel_launch(void* const* d_in, const int* in_sizes, int n_in,
                              void* d_out, int out_size, void* d_ws, size_t ws_size,
                              hipStream_t stream) {
    const float* vecs    = (const float*)d_in[0];
    const int*   rows    = (const int*)  d_in[1];
    const int*   cols    = (const int*)  d_in[2];
    const float* vals    = (const float*)d_in[3];
    const float* W_neigh = (const float*)d_in[4];
    const float* W_self  = (const float*)d_in[5];
    const float* offset  = (const float*)d_in[6];
    const float* scale   = (const float*)d_in[7];
    float* out = (float*)d_out;
    const int E = in_sizes[1];

    // 1) zero output / accumulator: N*512 floats = 12.8M float4
    zero_out_kernel<<<(N_NODES * OUTW / 4) / 256, 256, 0, stream>>>((float4*)out);

    // 2) SpMM edge scatter: one wave per edge, 8 waves per block
    edge_scatter_kernel<<<(E + 7) / 8, 256, 0, stream>>>(vecs, rows, cols, vals, out, E);

    // 3) fused dual-GEMM (WMMA f32) + layer-norm + ReLU: one block per 16 rows
    gemm_norm_kernel<<<N_NODES / 16, 256, 0, stream>>>(vecs, W_neigh, W_self,
                                                       offset, scale, out);
}